// MCPBRNN_Generic_PETconstraint_Three_VariantOutputGate_BYPASSM0_10196252360854
// MI455X (gfx1250) — compile-verified
//
#include <hip/hip_runtime.h>
#include <hip/hip_bf16.h>
#include <math.h>

#define NB       20000
#define NT       1024
#define SPIN     365
#define TRAINLEN 15000
#define ML_C     2.9086f
#define SL_C     1.898f
#define LOG2E_F  1.4426950408889634f

typedef int v4i __attribute__((ext_vector_type(4)));

// ---- CDNA5 async global->LDS path (guarded; fallback = plain copy) ----
#if __has_builtin(__builtin_amdgcn_global_load_async_to_lds_b128)
#define HAVE_ASYNC_B128 1
#else
#define HAVE_ASYNC_B128 0
#endif
#if __has_builtin(__builtin_amdgcn_s_wait_asynccnt)
#define WAIT_ASYNC() __builtin_amdgcn_s_wait_asynccnt(0)
#else
#define WAIT_ASYNC() asm volatile("s_wait_asynccnt 0" ::: "memory")
#endif

__device__ __forceinline__ float sigf(float x) {
    // sigmoid via v_exp_f32 + v_rcp_f32 (short critical path)
    float e = __builtin_amdgcn_exp2f(-x * LOG2E_F);
    return __builtin_amdgcn_rcpf(1.0f + e);
}

__global__ __launch_bounds__(NT) void mcpbrnn_kernel(
    const float* __restrict__ x,        // [NB,1,2] interleaved (u1,u2)
    const float* __restrict__ y_obs,    // [NB,1]
    const float* __restrict__ p_mean, const float* __restrict__ p_std,
    const int*   __restrict__ time_lag_p,
    const float* w_r_yom, const float* w_r_yom_fp, const float* w_r_yom_gw,
    const float* w_r_ylm, const float* w_r_yfm,
    const float* w_b1_yom, const float* w_b1_yom_gw, const float* w_b1_yom_fp,
    const float* w_b2_ylm, const float* theltaC,
    const float* b0_yom, const float* b0_yom_gw, const float* b0_yom_fp,
    const float* b0_ylm,
    float* __restrict__ out)
{
    __shared__ __align__(16) float sX[2 * NB];   // 160000 B : (u1,u2) pairs
    __shared__ float sOL[NB];                    //  80000 B : ol[b] (c-independent)
    __shared__ float sC[NB];                     //  80000 B : c trajectory (pre-state)
    __shared__ double sRed[64];                  //    512 B : reduction partials
    __shared__ float sObs;                       //      4 B

    const int tid = threadIdx.x;

    // ---- scalar parameter prep (every thread; trivial) ----
    const float e_yom = expf(w_r_yom[0]);
    const float e_gw  = expf(w_r_yom_gw[0]);
    const float e_ylm = expf(w_r_ylm[0]);
    const float e_yfm = expf(w_r_yfm[0]);
    const float e_fp  = expf(w_r_yom_fp[0]);
    const float inv_d = 1.0f / (e_yom + e_gw + e_ylm + e_yfm + e_fp);
    const float oo1   = e_yom * inv_d;
    const float oogw1 = e_gw  * inv_d;
    const float oofp1 = e_fp  * inv_d;
    const float ol1   = e_ylm * inv_d;
    const float expC  = expf(theltaC[0]);
    const float mo = p_mean[0], so = p_std[0];
    const float inv_so = 1.0f / so;
    const float kA = w_b1_yom[0]    * inv_so, dA = b0_yom[0]    - mo * kA;
    const float kG = w_b1_yom_gw[0] * inv_so, dG = b0_yom_gw[0] - mo * kG;
    const float kF = w_b1_yom_fp[0] * inv_so, dF = b0_yom_fp[0] - mo * kF;
    const float wol = w_b2_ylm[0], bol = b0_ylm[0];
    const int   tl = time_lag_p[0];

    // ---- Phase 1a: stage x[] (160 KB) into LDS via async global->LDS ----
#if HAVE_ASYNC_B128
    {
        const v4i* xg = (const v4i*)x;
        v4i* xl = (v4i*)sX;
        for (int i = tid; i < (2 * NB) / 4; i += NT) {
            __builtin_amdgcn_global_load_async_to_lds_b128(
                (v4i*)(xg + i), (v4i*)(xl + i), 0, 0);
        }
    }
#else
    for (int i = tid; i < (2 * NB) / 4; i += NT)
        ((float4*)sX)[i] = ((const float4*)x)[i];
#endif

    // ---- Phase 1b: ol[b] = ol1 * sig(b0_ylm + ((u2-ML)/SL)*w) — c-independent ----
    for (int b = tid; b < NB; b += NT) {
        float u2 = x[2 * b + 1];
        sOL[b] = ol1 * sigf(bol + ((u2 - ML_C) * (1.0f / SL_C)) * wol);
    }

    // ---- Phase 1c: obsstd = std(y_obs[365:15000], ddof=1), double accumulation ----
    {
        double s = 0.0, q = 0.0;
        for (int i = SPIN + tid; i < TRAINLEN; i += NT) {
            double v = (double)y_obs[i];
            s += v; q += v * v;
        }
        for (int off = 16; off > 0; off >>= 1) {
            s += __shfl_down(s, off, 32);
            q += __shfl_down(q, off, 32);
        }
        const int wave = tid >> 5, lane = tid & 31;
        if (lane == 0) { sRed[wave] = s; sRed[32 + wave] = q; }
    }

    WAIT_ASYNC();         // own wave's async->LDS transfers done
    __syncthreads();      // all waves' LDS writes visible

    if (tid == 0) {
        double S = 0.0, Q = 0.0;
        for (int w = 0; w < NT / 32; ++w) { S += sRed[w]; Q += sRed[32 + w]; }
        const double n = (double)(TRAINLEN - SPIN);
        double var = (Q - S * S / n) / (n - 1.0);
        sObs = (float)sqrt(var);
    }

    // ---- Phase 2: the inherently serial recurrence (one thread, LDS only) ----
    if (tid == 0) {
        float c0 = 0.0f;
        for (int b = 0; b < NB; ++b) {
            const float u1  = sX[2 * b];
            const float u2  = sX[2 * b + 1];
            const float olv = sOL[b];
            sC[b] = c0;                                   // pre-step state
            const float s1 = sigf(dA + kA * c0);
            const float s2 = sigf(dG + kG * c0);
            const float s3 = sigf(dF + kF * c0);
            const float oo   = oo1   * s1;
            const float oogw = oogw1 * s2;
            const float oofp = oofp1 * s3;
            const float qv  = u2 * __builtin_amdgcn_rcpf(c0);
            const float olc = (c0 > 0.0f) ? fminf(olv, qv) : olv;  // ol - relu(ol - u2/c0)
            const float f   = 1.0f - oo - oofp - oogw - olc;
            const float px  = fmaxf(u1 + c0 - expC, 0.0f);
            const float c1  = f * c0 + (u1 - px);
            c0 = (b >= tl) ? c1 : c0;
        }
    }
    __syncthreads();

    // ---- Phase 3: reconstruct all 16 outputs in parallel from (sC, sX, sOL) ----
    const float obsstd = sObs;
    for (int b = tid; b < NB; b += NT) {
        const float c0  = sC[b];
        const float u1  = sX[2 * b];
        const float u2  = sX[2 * b + 1];
        const float olv = sOL[b];
        const float px = fmaxf(u1 + c0 - expC, 0.0f);
        const float ib = (u1 > 0.0f) ? px * __builtin_amdgcn_rcpf(u1) : 0.0f;
        const float s1 = sigf(dA + kA * c0);
        const float s2 = sigf(dG + kG * c0);
        const float s3 = sigf(dF + kF * c0);
        const float oo   = oo1   * s1;
        const float oogw = oogw1 * s2;
        const float oofp = oofp1 * s3;
        const float qv  = u2 * __builtin_amdgcn_rcpf(c0);
        const float olc = (c0 > 0.0f) ? fminf(olv, qv) : olv;
        const float f   = 1.0f - oo - oofp - oogw - olc;
        const bool act = (b >= tl);
        const float h = oo * c0 + px;
        out[ 0 * NB + b] = act ? h           : 0.0f;  // h
        out[ 1 * NB + b] = act ? (oofp * c0) : 0.0f;  // hfp
        out[ 2 * NB + b] = act ? c0          : 0.0f;  // c
        out[ 3 * NB + b] = act ? (olv * c0)  : 0.0f;  // l
        out[ 4 * NB + b] = act ? (olc * c0)  : 0.0f;  // lc
        out[ 5 * NB + b] = act ? px          : 0.0f;  // bp
        out[ 6 * NB + b] = act ? (oogw * c0) : 0.0f;  // gw
        out[ 7 * NB + b] = act ? ib          : 0.0f;  // ib
        out[ 8 * NB + b] = act ? oo          : 0.0f;  // oo
        out[ 9 * NB + b] = act ? oofp        : 0.0f;  // oofp
        out[10 * NB + b] = act ? olv         : 0.0f;  // ol
        out[11 * NB + b] = act ? olc         : 0.0f;  // olc
        out[12 * NB + b] = act ? f           : 0.0f;  // f
        out[13 * NB + b] = act ? oogw        : 0.0f;  // oogw
        const float ob = act ? obsstd : 0.0f;
        out[14 * NB + 2 * b]     = act ? h : 0.0f;    // h_nout[:,0]
        out[14 * NB + 2 * b + 1] = ob;                // h_nout[:,1]
        out[16 * NB + b] = ob;                        // obs_std
    }
}

extern "C" void kernel_launch(void* const* d_in, const int* in_sizes, int n_in,
                              void* d_out, int out_size, void* d_ws, size_t ws_size,
                              hipStream_t stream) {
    (void)in_sizes; (void)n_in; (void)out_size; (void)d_ws; (void)ws_size;
    const float* x        = (const float*)d_in[0];
    const float* y_obs    = (const float*)d_in[1];
    const float* p_mean   = (const float*)d_in[2];
    const float* p_std    = (const float*)d_in[3];
    /* d_in[4] = epoch (unused) */
    const int*   time_lag = (const int*)d_in[5];
    const float* w_r_yom     = (const float*)d_in[6];
    const float* w_r_yom_fp  = (const float*)d_in[7];
    const float* w_r_yom_gw  = (const float*)d_in[8];
    const float* w_r_ylm     = (const float*)d_in[9];
    const float* w_r_yfm     = (const float*)d_in[10];
    const float* w_b1_yom    = (const float*)d_in[11];
    const float* w_b1_yom_gw = (const float*)d_in[12];
    const float* w_b1_yom_fp = (const float*)d_in[13];
    const float* w_b2_ylm    = (const float*)d_in[14];
    const float* theltaC     = (const float*)d_in[15];
    const float* b0_yom      = (const float*)d_in[16];
    const float* b0_yom_gw   = (const float*)d_in[17];
    const float* b0_yom_fp   = (const float*)d_in[18];
    const float* b0_ylm      = (const float*)d_in[19];
    float* out = (float*)d_out;

    hipLaunchKernelGGL(mcpbrnn_kernel, dim3(1), dim3(NT), 0, stream,
                       x, y_obs, p_mean, p_std, time_lag,
                       w_r_yom, w_r_yom_fp, w_r_yom_gw, w_r_ylm, w_r_yfm,
                       w_b1_yom, w_b1_yom_gw, w_b1_yom_fp, w_b2_ylm, theltaC,
                       b0_yom, b0_yom_gw, b0_yom_fp, b0_ylm, out);
}